// PtModule_76166950027887
// MI455X (gfx1250) — compile-verified
//
#include <hip/hip_runtime.h>

// out = (x + 2) * 3 - 1, elementwise over 8192*16384 fp32.
// HBM-bound: ~1.07 GB traffic -> ~46 us floor at 23.3 TB/s. Compute is noise.
// This revision: block-uniform full-tile fast path so each wave issues 4
// back-to-back global_load_b128 (NT) before any s_wait_loadcnt -> 4x the
// per-wave memory-level parallelism vs the per-access-predicated version.

typedef float v4f __attribute__((ext_vector_type(4)));

constexpr int BLOCK = 256;          // 8 wave32 waves per block
constexpr int VEC_PER_THREAD = 4;   // 4 x 16B = 64B per thread

__device__ __forceinline__ v4f affine(v4f v) {
    return (v + 2.0f) * 3.0f - 1.0f;   // same op order as reference
}

__global__ __launch_bounds__(BLOCK)
void affine_stream_kernel(const float* __restrict__ in,
                          float* __restrict__ out,
                          long long n) {
    const long long n4        = n >> 2;                        // # float4 elems
    const long long tile      = (long long)BLOCK * VEC_PER_THREAD;
    const long long tile_base = (long long)blockIdx.x * tile;

    const v4f* __restrict__ in4  = (const v4f*)in;
    v4f* __restrict__       out4 = (v4f*)out;

    if (tile_base + tile <= n4) {
        // ---- Fast path: whole tile in range (block-uniform, no lane checks).
        const long long i0 = tile_base + threadIdx.x;
        // Issue all 4 loads before any use -> 4 b128 NT loads in flight.
        v4f a = __builtin_nontemporal_load(in4 + i0 + 0 * BLOCK);
        v4f b = __builtin_nontemporal_load(in4 + i0 + 1 * BLOCK);
        v4f c = __builtin_nontemporal_load(in4 + i0 + 2 * BLOCK);
        v4f d = __builtin_nontemporal_load(in4 + i0 + 3 * BLOCK);
        __builtin_nontemporal_store(affine(a), out4 + i0 + 0 * BLOCK);
        __builtin_nontemporal_store(affine(b), out4 + i0 + 1 * BLOCK);
        __builtin_nontemporal_store(affine(c), out4 + i0 + 2 * BLOCK);
        __builtin_nontemporal_store(affine(d), out4 + i0 + 3 * BLOCK);
    } else {
        // ---- Tail path: per-access predication (last block only).
#pragma unroll
        for (int k = 0; k < VEC_PER_THREAD; ++k) {
            long long i = tile_base + threadIdx.x + (long long)k * BLOCK;
            if (i < n4) {
                v4f v = __builtin_nontemporal_load(in4 + i);
                __builtin_nontemporal_store(affine(v), out4 + i);
            }
        }
        // Scalar remainder for n % 4 (at most 3 elements).
        const long long tail_start = n4 << 2;
        const long long ti = tail_start + threadIdx.x;
        if (ti < n) {
            float x = __builtin_nontemporal_load(in + ti);
            __builtin_nontemporal_store((x + 2.0f) * 3.0f - 1.0f, out + ti);
        }
    }
}

extern "C" void kernel_launch(void* const* d_in, const int* in_sizes, int n_in,
                              void* d_out, int out_size, void* d_ws, size_t ws_size,
                              hipStream_t stream) {
    (void)n_in; (void)d_ws; (void)ws_size; (void)out_size;

    const float* in  = (const float*)d_in[0];
    float*       out = (float*)d_out;
    const long long n = (long long)in_sizes[0];

    const long long tile = (long long)BLOCK * VEC_PER_THREAD;
    const long long n4   = n >> 2;
    long long blocks = (n4 + tile - 1) / tile;
    if (blocks < 1) blocks = 1;                 // tiny-n: tail path covers it

    affine_stream_kernel<<<(unsigned)blocks, BLOCK, 0, stream>>>(in, out, n);
}